// Encoder_82566451298883
// MI455X (gfx1250) — compile-verified
//
#include <hip/hip_runtime.h>
#include <hip/hip_bf16.h>

typedef __attribute__((ext_vector_type(2))) float v2f;
typedef __attribute__((ext_vector_type(8))) float v8f;

// ---------------------------------------------------------------------------
// Degree / normalization kernels
// ---------------------------------------------------------------------------
__global__ void k_init_deg(float* __restrict__ deg, int n) {
    int i = blockIdx.x * blockDim.x + threadIdx.x;
    if (i < n) deg[i] = 1.0f;  // self loop
}

__global__ void k_count_deg(const int* __restrict__ dst, float* __restrict__ deg, int e) {
    int i = blockIdx.x * blockDim.x + threadIdx.x;
    if (i < e) atomicAdd(&deg[dst[i]], 1.0f);
}

__global__ void k_finalize_dis(float* __restrict__ dis, int n) {
    int i = blockIdx.x * blockDim.x + threadIdx.x;
    if (i < n) dis[i] = rsqrtf(dis[i]);  // deg >= 1 always (self loop)
}

// ---------------------------------------------------------------------------
// Fused dense GEMM + self-loop/bias epilogue.
//   H[M x FOUT]   = (RELU? max(X,0):X) @ W
//   OUT[M x FOUT] = H * dis^2 + bias        (accumulator init for scatter)
//
// V_WMMA_F32_16X16X4_F32; MT 16-row M-tiles per block, wave w owns N-tile w.
// sA rows padded to FIN+1 floats (odd stride) -> bank-conflict-free A frags.
// B fragments preloaded once and reused across all MT tiles.
//
// f32 16x16x4 fragment layouts (ISA 7.12.2):
//   A (16x4): lanes 0-15 -> M=lane, v0=K0,v1=K1 ; lanes 16-31 -> v0=K2,v1=K3
//   B (4x16): lanes 0-15 -> N=lane, v0=K0,v1=K1 ; lanes 16-31 -> v0=K2,v1=K3
//   C/D:      vgpr r: lanes 0-15 -> (M=r, N=lane); lanes 16-31 -> (M=r+8, N=lane-16)
// ---------------------------------------------------------------------------
template <int FIN, int FOUT, int MT, bool RELU>
__global__ void k_gemm_wmma(const float* __restrict__ X, const float* __restrict__ W,
                            const float* __restrict__ dis, const float* __restrict__ bias,
                            float* __restrict__ H, float* __restrict__ OUT, int M) {
    constexpr int LDA = FIN + 1;                 // padded row stride
    __shared__ float sW[FIN * FOUT];
    __shared__ float sA[MT * 16 * LDA];

    const int tid  = threadIdx.x;
    const int nthr = 32 * (FOUT / 16);
    const int m0   = blockIdx.x * (MT * 16);

    for (int i = tid; i < FIN * FOUT; i += nthr) sW[i] = W[i];
    for (int i = tid; i < MT * 16 * FIN; i += nthr) {
        int r = i / FIN, c = i - r * FIN;
        int row = m0 + r; if (row >= M) row = M - 1;   // clamp; stores guarded
        float v = X[(size_t)row * FIN + c];
        sA[r * LDA + c] = RELU ? fmaxf(v, 0.0f) : v;
    }
    __syncthreads();

    const int wave = tid >> 5;
    const int lane = tid & 31;
    const int half = lane >> 4;    // 0: K pair {k,k+1}; 1: {k+2,k+3}
    const int l16  = lane & 15;
    const int n0   = wave * 16;

    // B fragments for this wave's N-tile, reused across all MT tiles
    v2f bf[FIN / 4];
    #pragma unroll
    for (int kk = 0; kk < FIN / 4; ++kk) {
        int k = kk * 4 + 2 * half;
        bf[kk].x = sW[k * FOUT + n0 + l16];
        bf[kk].y = sW[(k + 1) * FOUT + n0 + l16];
    }
    const float bcol = bias[n0 + l16];

    #pragma unroll
    for (int mt = 0; mt < MT; ++mt) {
        const float* At = &sA[mt * 16 * LDA];
        v8f c = {};
        #pragma unroll
        for (int kk = 0; kk < FIN / 4; ++kk) {
            int k = kk * 4 + 2 * half;
            v2f a;
            a.x = At[l16 * LDA + k];
            a.y = At[l16 * LDA + k + 1];
            // 8 args: (neg_a, A, neg_b, B, c_mod, C, reuse_a, reuse_b)
            c = __builtin_amdgcn_wmma_f32_16x16x4_f32(false, a, false, bf[kk],
                                                      (short)0, c, false, false);
        }
        #pragma unroll
        for (int r = 0; r < 8; ++r) {
            int row = m0 + mt * 16 + half * 8 + r;
            if (row < M) {
                float hv = c[r];
                float s  = dis[row];
                size_t o = (size_t)row * FOUT + n0 + l16;
                H[o]   = hv;
                OUT[o] = hv * (s * s) + bcol;
            }
        }
    }
}

// ---------------------------------------------------------------------------
// Edge scatter: out[dst, :] += h[src, :] * dis[src] * dis[dst]
// One thread per (edge, 4-feature chunk): float4 gather + 4 f32 atomics.
// CHUNKS (= F/4) is a template constant so edge/chunk decomposition is
// shift/mask (no integer divide in the hot loop); all indexing 32-bit
// (total work <= 51.2M items). Consecutive lanes share an edge, so the
// src/dst/dis loads coalesce to broadcasts; atomics resolve in the 192MB L2
// where the whole accumulator lives.
// ---------------------------------------------------------------------------
template <int CHUNKS>
__global__ void k_scatter(const float* __restrict__ h, const int* __restrict__ src,
                          const int* __restrict__ dst, const float* __restrict__ dis,
                          float* __restrict__ out, int e) {
    constexpr int F = CHUNKS * 4;
    const unsigned total = (unsigned)e * (unsigned)CHUNKS;
    const unsigned step  = gridDim.x * blockDim.x;
    for (unsigned idx = blockIdx.x * blockDim.x + threadIdx.x; idx < total; idx += step) {
        const unsigned ei = idx / CHUNKS;           // constant divisor -> shift
        const unsigned c  = idx - ei * CHUNKS;      // -> mask
        const int s = src[ei], d = dst[ei];
        const float norm = dis[s] * dis[d];
        const float4 v = *(const float4*)(h + (size_t)s * F + (c << 2));
        float* o = out + (size_t)d * F + (c << 2);
        atomicAdd(o + 0, v.x * norm);
        atomicAdd(o + 1, v.y * norm);
        atomicAdd(o + 2, v.z * norm);
        atomicAdd(o + 3, v.w * norm);
    }
}

// ---------------------------------------------------------------------------
// Launch
// ---------------------------------------------------------------------------
extern "C" void kernel_launch(void* const* d_in, const int* in_sizes, int n_in,
                              void* d_out, int out_size, void* d_ws, size_t ws_size,
                              hipStream_t stream) {
    const float* X  = (const float*)d_in[0];   // [N, 64]
    const int*   EI = (const int*)d_in[1];     // [2, E]
    const float* W1 = (const float*)d_in[2];   // [64, 128]
    const float* b1 = (const float*)d_in[3];   // [128]
    const float* W2 = (const float*)d_in[4];   // [128, 64]
    const float* b2 = (const float*)d_in[5];   // [64]
    float* out = (float*)d_out;                // [N, 64]

    const int FIN1 = 64;
    const int N = in_sizes[0] / FIN1;          // 100000
    const int E = in_sizes[1] / 2;             // 1600000
    const int* src = EI;
    const int* dst = EI + E;

    // workspace: dis[N] | bufH[N*128] | bufO[N*128]
    float* dis = (float*)d_ws;
    size_t off = ((size_t)N * sizeof(float) + 255) & ~(size_t)255;
    float* bufH = (float*)((char*)d_ws + off);
    float* bufO = bufH + (size_t)N * 128;

    const int T = 256;
    dim3 blkN((N + T - 1) / T), blkE((E + T - 1) / T);
    const int ntiles = (N + 15) / 16;          // 6250

    // degrees -> dis = rsqrt(deg)
    k_init_deg<<<blkN, T, 0, stream>>>(dis, N);
    k_count_deg<<<blkE, T, 0, stream>>>(dst, dis, E);
    k_finalize_dis<<<blkN, T, 0, stream>>>(dis, N);

    // layer 1: h = X @ W1 ; fused out1 init = h*dis^2 + b1 ; edge scatter
    k_gemm_wmma<64, 128, 4, false><<<(ntiles + 3) / 4, 256, 0, stream>>>(
        X, W1, dis, b1, bufH, bufO, N);
    k_scatter<32><<<8192, T, 0, stream>>>(bufH, src, dst, dis, bufO, E);

    // layer 2: h2 = relu(out1) @ W2 (into bufH) ; fused out init ; scatter
    k_gemm_wmma<128, 64, 2, true><<<(ntiles + 1) / 2, 128, 0, stream>>>(
        bufO, W2, dis, b2, bufH, out, N);
    k_scatter<16><<<8192, T, 0, stream>>>(bufH, src, dst, dis, out, E);
}